// MiniMaxMoEReference_10840497455873
// MI455X (gfx1250) — compile-verified
//
#include <hip/hip_runtime.h>
#include <hip/hip_bf16.h>
#include <math.h>
#include <stdint.h>

// Problem constants (from reference)
#define NE     64
#define TOPK   6
#define HID    2048
#define INTER  1024
#define RSCALE 1.0f

typedef __attribute__((ext_vector_type(2))) float v2f;
typedef __attribute__((ext_vector_type(8))) float v8f;

// 128 + 4 pad: row-to-row bank shift of 4 -> WMMA fragment ds reads hit 64 distinct banks.
// 132 is a multiple of 4 -> every row stays 16B aligned for b128 LDS writes.
#define TSTRIDE 132

__device__ __forceinline__ v8f wmma4(v2f a, v2f b, v8f c) {
  // D = A(16x4 f32) * B(4x16 f32) + C(16x16 f32)
  return __builtin_amdgcn_wmma_f32_16x16x4_f32(false, a, false, b, (short)0, c, false, false);
}

// CDNA5 async global->LDS DMA (ASYNCcnt-tracked, no VGPR staging).
// vdst = 32-bit LDS byte address, vaddr = 64-bit global address.
__device__ __forceinline__ void async_load_b128(uint32_t lds_addr, uint64_t gaddr) {
  asm volatile("global_load_async_to_lds_b128 %0, %1, off"
               :: "v"(lds_addr), "v"(gaddr)
               : "memory");
}
// Generic (flat) LDS pointers map to LDS as addr[31:0] (ISA 10.2 aperture rules).
__device__ __forceinline__ uint32_t lds_addr_of(const void* p) {
  return (uint32_t)(size_t)p;
}
#define WAIT_ASYNC(n) asm volatile("s_wait_asynccnt " #n ::: "memory")

// ---------------- Kernel 1: router (logits -> sigmoid -> +bias -> top-6 -> weights) --------
__global__ void moe_route(const float* __restrict__ x, const float* __restrict__ gw,
                          const float* __restrict__ bias, int* __restrict__ idx_out,
                          float* __restrict__ w_out) {
  __shared__ float part[256];
  __shared__ float logits[NE];
  const int t = threadIdx.x;
  const int e = t >> 2, q = t & 3;      // 4 threads per expert, 512 elems each
  const float4* gw4 = reinterpret_cast<const float4*>(gw + (size_t)e * HID + q * 512);
  const float4* x4  = reinterpret_cast<const float4*>(x + q * 512);
  float s = 0.f;
  for (int j = 0; j < 128; ++j) {
    float4 a = gw4[j], b = x4[j];
    s += a.x * b.x + a.y * b.y + a.z * b.z + a.w * b.w;
  }
  part[t] = s;
  __syncthreads();
  if (t < NE) logits[t] = part[4 * t] + part[4 * t + 1] + part[4 * t + 2] + part[4 * t + 3];
  __syncthreads();
  if (t == 0) {
    float sc[NE];
    bool taken[NE];
    for (int i = 0; i < NE; ++i) { sc[i] = 1.f / (1.f + expf(-logits[i])); taken[i] = false; }
    int sel[TOPK]; float selw[TOPK]; float ssum = 0.f;
    for (int kk = 0; kk < TOPK; ++kk) {
      int best = 0; float bv = -1e30f;
      for (int i = 0; i < NE; ++i) {
        if (!taken[i]) {
          float b = sc[i] + bias[i];
          if (b > bv) { bv = b; best = i; }   // strict '>' keeps earliest index on ties
        }
      }
      taken[best] = true; sel[kk] = best; selw[kk] = sc[best]; ssum += sc[best];
    }
    float inv = RSCALE / (ssum + 1e-20f);
    for (int kk = 0; kk < TOPK; ++kk) { idx_out[kk] = sel[kk]; w_out[kk] = selw[kk] * inv; }
  }
}

// ---------------- Kernel 2: h[k,i] = silu(w1[e]@x) * (w3[e]@x) ------------------------------
// grid.x = TOPK * (INTER/16); block owns 16 rows of one expert; 8 waves split K.
// Double-buffered 16x128 tiles filled by async global->LDS loads; next chunk's DMA is in
// flight while current chunk's WMMAs execute. Each wave issues 4 async ops per chunk, so
// s_wait_asynccnt 4 retires the current buffer while the next stays outstanding.
__global__ void moe_gate_up(const float* __restrict__ x, const float* __restrict__ w1,
                            const float* __restrict__ w3, const int* __restrict__ idx,
                            float* __restrict__ hbuf) {
  __shared__ __align__(16) float sW1[2][16 * TSTRIDE];
  __shared__ __align__(16) float sW3[2][16 * TSTRIDE];
  __shared__ float sX[2][128];
  __shared__ float red1[8][16];
  __shared__ float red3[8][16];

  const int t = threadIdx.x;
  const int kk = blockIdx.x >> 6;                 // which selected expert (0..5)
  const int row_base = (blockIdx.x & 63) * 16;    // rows of INTER
  const int e = idx[kk];
  const float* w1b = w1 + ((size_t)e * INTER + row_base) * HID;
  const float* w3b = w3 + ((size_t)e * INTER + row_base) * HID;

  const int lane = t & 31;
  const int wv   = t >> 5;
  const int row  = lane & 15;
  const int koff = (lane < 16) ? 0 : 2;           // WMMA f32 A/B fragment K selector

  // Per-thread slice of the 16x128 tile copy: 2 x b128 per matrix.
  const int l0 = t;                 // linear float4 index, rep 0
  const int fr0 = l0 >> 5, fc0 = (l0 & 31) << 2;
  const int l1 = 256 + t;           // rep 1
  const int fr1 = l1 >> 5, fc1 = (l1 & 31) << 2;

  v8f acc1 = {}; v8f acc3 = {};
  const int NCH = HID / 128;

  // prologue: issue chunk 0 into buffer 0
  {
    async_load_b128(lds_addr_of(&sW1[0][fr0 * TSTRIDE + fc0]), (uint64_t)(size_t)(w1b + (size_t)fr0 * HID + fc0));
    async_load_b128(lds_addr_of(&sW1[0][fr1 * TSTRIDE + fc1]), (uint64_t)(size_t)(w1b + (size_t)fr1 * HID + fc1));
    async_load_b128(lds_addr_of(&sW3[0][fr0 * TSTRIDE + fc0]), (uint64_t)(size_t)(w3b + (size_t)fr0 * HID + fc0));
    async_load_b128(lds_addr_of(&sW3[0][fr1 * TSTRIDE + fc1]), (uint64_t)(size_t)(w3b + (size_t)fr1 * HID + fc1));
    if (t < 128) sX[0][t] = x[t];
  }

  for (int ch = 0; ch < NCH; ++ch) {
    const int cur = ch & 1, nxt = cur ^ 1;
    if (ch + 1 < NCH) {                           // issue next chunk's DMA into other buffer
      const int kn = (ch + 1) * 128;
      async_load_b128(lds_addr_of(&sW1[nxt][fr0 * TSTRIDE + fc0]), (uint64_t)(size_t)(w1b + (size_t)fr0 * HID + kn + fc0));
      async_load_b128(lds_addr_of(&sW1[nxt][fr1 * TSTRIDE + fc1]), (uint64_t)(size_t)(w1b + (size_t)fr1 * HID + kn + fc1));
      async_load_b128(lds_addr_of(&sW3[nxt][fr0 * TSTRIDE + fc0]), (uint64_t)(size_t)(w3b + (size_t)fr0 * HID + kn + fc0));
      async_load_b128(lds_addr_of(&sW3[nxt][fr1 * TSTRIDE + fc1]), (uint64_t)(size_t)(w3b + (size_t)fr1 * HID + kn + fc1));
      if (t < 128) sX[nxt][t] = x[kn + t];
      WAIT_ASYNC(0x4);                            // current buffer's 4 ops retired
    } else {
      WAIT_ASYNC(0x0);
    }
    __syncthreads();                              // all waves' current-chunk data visible
#pragma unroll
    for (int u = 0; u < 4; ++u) {
      const int k = wv * 16 + u * 4 + koff;
      v2f b;  b.x  = sX[cur][k];                      b.y  = sX[cur][k + 1];   // broadcast over N
      v2f a1; a1.x = sW1[cur][row * TSTRIDE + k];     a1.y = sW1[cur][row * TSTRIDE + k + 1];
      acc1 = wmma4(a1, b, acc1);
      v2f a3; a3.x = sW3[cur][row * TSTRIDE + k];     a3.y = sW3[cur][row * TSTRIDE + k + 1];
      acc3 = wmma4(a3, b, acc3);
    }
    __syncthreads();                              // reads done -> buffer reusable
  }

  // D columns are identical; row m<8 lives in VGPR m on lanes 0-15, row m+8 on lanes 16-31.
  if (lane == 0) {
#pragma unroll
    for (int v = 0; v < 8; ++v) { red1[wv][v] = acc1[v]; red3[wv][v] = acc3[v]; }
  }
  if (lane == 16) {
#pragma unroll
    for (int v = 0; v < 8; ++v) { red1[wv][8 + v] = acc1[v]; red3[wv][8 + v] = acc3[v]; }
  }
  __syncthreads();
  if (t < 16) {
    float g = 0.f, u = 0.f;
#pragma unroll
    for (int w = 0; w < 8; ++w) { g += red1[w][t]; u += red3[w][t]; }
    float sg = g / (1.f + expf(-g));              // silu(g)
    hbuf[kk * INTER + row_base + t] = sg * u;
  }
}

// ---------------- Kernel 3: out[h] = sum_k w_k * (w2[e_k] @ h_k) ----------------------------
// grid.x = HID/16; expert x chunk loop flattened so the async double-buffer pipelines across
// expert boundaries. Routing weight folded into B; accumulation over all experts stays in the
// WMMA accumulator -> no atomics, every output element fully overwritten.
__global__ void moe_down(const float* __restrict__ w2, const int* __restrict__ idx,
                         const float* __restrict__ wts, const float* __restrict__ hbuf,
                         float* __restrict__ out) {
  __shared__ __align__(16) float sW[2][16 * TSTRIDE];
  __shared__ float sH[2][128];
  __shared__ float red[8][16];

  const int t = threadIdx.x;
  const int row_base = blockIdx.x * 16;           // rows of HIDDEN
  const int lane = t & 31;
  const int wv   = t >> 5;
  const int row  = lane & 15;
  const int koff = (lane < 16) ? 0 : 2;

  const int l0 = t;
  const int fr0 = l0 >> 5, fc0 = (l0 & 31) << 2;
  const int l1 = 256 + t;
  const int fr1 = l1 >> 5, fc1 = (l1 & 31) << 2;

  v8f acc = {};
  const int NCH = INTER / 128;
  const int NITER = TOPK * NCH;

  // prologue: (expert 0, chunk 0) into buffer 0
  {
    const int e0 = idx[0];
    const float wk0 = wts[0];
    const float* wb = w2 + ((size_t)e0 * HID + row_base) * INTER;
    async_load_b128(lds_addr_of(&sW[0][fr0 * TSTRIDE + fc0]), (uint64_t)(size_t)(wb + (size_t)fr0 * INTER + fc0));
    async_load_b128(lds_addr_of(&sW[0][fr1 * TSTRIDE + fc1]), (uint64_t)(size_t)(wb + (size_t)fr1 * INTER + fc1));
    if (t < 128) sH[0][t] = wk0 * hbuf[t];
  }

  for (int ci = 0; ci < NITER; ++ci) {
    const int cur = ci & 1, nxt = cur ^ 1;
    if (ci + 1 < NITER) {                         // issue next (expert, chunk) DMA
      const int kk = (ci + 1) / NCH;
      const int k0 = ((ci + 1) % NCH) * 128;
      const int en = idx[kk];
      const float wk = wts[kk];
      const float* wb = w2 + ((size_t)en * HID + row_base) * INTER + k0;
      async_load_b128(lds_addr_of(&sW[nxt][fr0 * TSTRIDE + fc0]), (uint64_t)(size_t)(wb + (size_t)fr0 * INTER + fc0));
      async_load_b128(lds_addr_of(&sW[nxt][fr1 * TSTRIDE + fc1]), (uint64_t)(size_t)(wb + (size_t)fr1 * INTER + fc1));
      if (t < 128) sH[nxt][t] = wk * hbuf[kk * INTER + k0 + t];
      WAIT_ASYNC(0x2);                            // current buffer's 2 ops retired
    } else {
      WAIT_ASYNC(0x0);
    }
    __syncthreads();
#pragma unroll
    for (int u = 0; u < 4; ++u) {
      const int k = wv * 16 + u * 4 + koff;
      v2f b; b.x = sH[cur][k];                   b.y = sH[cur][k + 1];
      v2f a; a.x = sW[cur][row * TSTRIDE + k];   a.y = sW[cur][row * TSTRIDE + k + 1];
      acc = wmma4(a, b, acc);
    }
    __syncthreads();
  }

  if (lane == 0) {
#pragma unroll
    for (int v = 0; v < 8; ++v) red[wv][v] = acc[v];
  }
  if (lane == 16) {
#pragma unroll
    for (int v = 0; v < 8; ++v) red[wv][8 + v] = acc[v];
  }
  __syncthreads();
  if (t < 16) {
    float s = 0.f;
#pragma unroll
    for (int w = 0; w < 8; ++w) s += red[w][t];
    out[row_base + t] = s;
  }
}

// ---------------- Host launch ----------------------------------------------------------------
extern "C" void kernel_launch(void* const* d_in, const int* in_sizes, int n_in,
                              void* d_out, int out_size, void* d_ws, size_t ws_size,
                              hipStream_t stream) {
  const float* x    = (const float*)d_in[0];
  const float* gw   = (const float*)d_in[1];
  const float* bias = (const float*)d_in[2];
  const float* w1   = (const float*)d_in[3];
  const float* w2   = (const float*)d_in[4];
  const float* w3   = (const float*)d_in[5];
  float* out = (float*)d_out;

  char* ws = (char*)d_ws;
  int*   ws_idx = (int*)ws;                 // [TOPK]
  float* ws_w   = (float*)(ws + 64);        // [TOPK]
  float* ws_h   = (float*)(ws + 256);       // [TOPK][INTER]

  moe_route<<<1, 256, 0, stream>>>(x, gw, bias, ws_idx, ws_w);
  moe_gate_up<<<TOPK * (INTER / 16), 256, 0, stream>>>(x, w1, w3, ws_idx, ws_h);
  moe_down<<<HID / 16, 256, 0, stream>>>(w2, ws_idx, ws_w, ws_h, out);
}